// ContrastiveLossWithAttention_13426067767692
// MI455X (gfx1250) — compile-verified
//
#include <hip/hip_runtime.h>
#include <hip/hip_bf16.h>
#include <math.h>

// Problem sizes (fixed by the reference).
#define B_ 16
#define N_ 2048
#define M_ 2048

typedef float v2f __attribute__((ext_vector_type(2)));
typedef float v4f __attribute__((ext_vector_type(4)));
typedef float v8f __attribute__((ext_vector_type(8)));

// D(16x16,f32) = A(16x4,f32) x B(4x16,f32) + C.
// With B == ones, every column of D holds the accumulated row-sum of A's rows.
// Column *order* inside A is irrelevant when B is all-ones, which lets us feed
// A-tiles built directly from contiguous b128 loads (low lanes: cols j0..j0+3,
// high lanes: cols j0+4..j0+7).
__device__ __forceinline__ v8f wmma_rowacc(v2f a, v2f b_ones, v8f c) {
    return __builtin_amdgcn_wmma_f32_16x16x4_f32(
        /*neg_a=*/false, a, /*neg_b=*/false, b_ones,
        /*c_mod=*/(short)0, c, /*reuse_a=*/false, /*reuse_b=*/false);
}

// Reduce four independent values over each 16-lane half-wave group using a
// packed butterfly: 8 bpermutes + 3 selects instead of 16 bpermutes.
// Result: lane with group-bits (b3,b2) holds the total of value k = b3 + 2*b2
// (replicated over bits 1,0).
__device__ __forceinline__ float quad_group_reduce(float v0, float v1,
                                                   float v2, float v3,
                                                   int lane) {
    v0 += __shfl_xor(v0, 8, 32);
    v1 += __shfl_xor(v1, 8, 32);
    v2 += __shfl_xor(v2, 8, 32);
    v3 += __shfl_xor(v3, 8, 32);
    const bool b3 = (lane & 8) != 0;
    float a = b3 ? v1 : v0;
    float b = b3 ? v3 : v2;
    a += __shfl_xor(a, 4, 32);
    b += __shfl_xor(b, 4, 32);
    float c = ((lane & 4) != 0) ? b : a;
    c += __shfl_xor(c, 2, 32);
    c += __shfl_xor(c, 1, 32);
    return c;
}

__device__ __forceinline__ float block_reduce_sum(float v, float* sdata) {
    const int tid = threadIdx.x;
    sdata[tid] = v;
    __syncthreads();
    for (int s = blockDim.x >> 1; s > 0; s >>= 1) {
        if (tid < s) sdata[tid] += sdata[tid + s];
        __syncthreads();
    }
    return sdata[0];
}

// ---------------------------------------------------------------------------
// Pass 1: row_gt/row_cnt (row sums of gt_pred/gt) via WMMA,
//         col_gt/col_cnt (col sums) via packed shuffle + LDS + global atomics.
// Grid: (N/128, B), 256 threads = 8 waves; wave w owns rows blk*128+w*16 .. +15.
// ---------------------------------------------------------------------------
__global__ __launch_bounds__(256)
void k_stats(const float* __restrict__ pred, const float* __restrict__ gt,
             const int* __restrict__ src_ns, const int* __restrict__ tgt_ns,
             float* __restrict__ row_gt, float* __restrict__ row_cnt,
             float* __restrict__ col_gt, float* __restrict__ col_cnt) {
    __shared__ float cacc_gp[M_];
    __shared__ float cacc_g[M_];
    const int b   = blockIdx.y;
    const int tid = threadIdx.x;
    for (int i = tid; i < M_; i += blockDim.x) { cacc_gp[i] = 0.f; cacc_g[i] = 0.f; }
    __syncthreads();

    const int wave = tid >> 5, lane = tid & 31;
    const int rowbase = blockIdx.x * 128 + wave * 16;
    const int row     = rowbase + (lane & 15);        // A-layout: lane -> row
    const int src = src_ns[b], tgt = tgt_ns[b];
    const float rowv = (row < src) ? 1.f : 0.f;
    const size_t base = ((size_t)b * N_ + row) * (size_t)M_;
    const int cshift = (lane >> 4) << 2;              // high lanes: cols +4..+7

    v8f c_gp = {}; v8f c_g = {};
    const v2f ones = {1.f, 1.f};

    for (int j0 = 0; j0 < M_; j0 += 8) {
        const int cb = j0 + cshift;                   // 16B aligned
        v4f pv = *(const v4f*)(pred + base + cb);
        v4f gv = *(const v4f*)(gt   + base + cb);

        float p[4], g[4], gp[4];
        #pragma unroll
        for (int k = 0; k < 4; ++k) {
            const float m = (cb + k < tgt) ? rowv : 0.f;
            p[k]  = fminf(fmaxf(pv[k], 0.f), 1.f) * m;
            g[k]  = gv[k] * m;
            gp[k] = p[k] * g[k];
        }

        // Row sums via matrix engine (two column-subsets per b128 load).
        c_gp = wmma_rowacc((v2f){gp[0], gp[1]}, ones, c_gp);
        c_gp = wmma_rowacc((v2f){gp[2], gp[3]}, ones, c_gp);
        c_g  = wmma_rowacc((v2f){g[0],  g[1]},  ones, c_g);
        c_g  = wmma_rowacc((v2f){g[2],  g[3]},  ones, c_g);

        // Column sums over this wave's 16 rows.
        const float sgp = quad_group_reduce(gp[0], gp[1], gp[2], gp[3], lane);
        const float sg  = quad_group_reduce(g[0],  g[1],  g[2],  g[3],  lane);
        if ((lane & 3) == 0) {
            const int vk  = ((lane >> 3) & 1) | (((lane >> 2) & 1) << 1);
            const int col = cb + vk;
            atomicAdd(&cacc_gp[col], sgp);
            atomicAdd(&cacc_g[col],  sg);
        }
    }

    // C-layout: lane 0 holds rows rowbase..+7 (VGPR k = row k), lane 16 rows +8..+15.
    if (lane == 0 || lane == 16) {
        const size_t ro = (size_t)b * N_ + rowbase + ((lane >> 4) << 3);
        #pragma unroll
        for (int k = 0; k < 8; ++k) {
            row_gt[ro + k]  = c_gp[k];
            row_cnt[ro + k] = c_g[k];
        }
    }

    __syncthreads();
    for (int i = tid; i < M_; i += blockDim.x) {
        atomicAdd(&col_gt[(size_t)b * M_ + i],  cacc_gp[i]);
        atomicAdd(&col_cnt[(size_t)b * M_ + i], cacc_g[i]);
    }
}

// ---------------------------------------------------------------------------
// Pass 2: src_neg/src_pos (row sums via WMMA), tgt_neg (col sums via LDS).
// ---------------------------------------------------------------------------
__global__ __launch_bounds__(256)
void k_terms(const float* __restrict__ pred, const float* __restrict__ gt,
             const int* __restrict__ src_ns, const int* __restrict__ tgt_ns,
             const float* __restrict__ beta_p,
             const float* __restrict__ row_gt, const float* __restrict__ row_cnt,
             const float* __restrict__ col_gt, const float* __restrict__ col_cnt,
             float* __restrict__ src_neg, float* __restrict__ src_pos,
             float* __restrict__ tgt_neg) {
    __shared__ float cacc_tn[M_];
    const int b   = blockIdx.y;
    const int tid = threadIdx.x;
    for (int i = tid; i < M_; i += blockDim.x) cacc_tn[i] = 0.f;
    __syncthreads();

    const int wave = tid >> 5, lane = tid & 31;
    const int rowbase = blockIdx.x * 128 + wave * 16;
    const int row     = rowbase + (lane & 15);
    const int src = src_ns[b], tgt = tgt_ns[b];
    const float beta = beta_p[0];
    const float rowv = (row < src) ? 1.f : 0.f;
    const float rg = row_gt[(size_t)b * N_ + row];
    const float rc = row_cnt[(size_t)b * N_ + row];
    const size_t base  = ((size_t)b * N_ + row) * (size_t)M_;
    const size_t cstat = (size_t)b * M_;
    const int cshift = (lane >> 4) << 2;

    v8f c_sn = {}; v8f c_sp = {};
    const v2f ones = {1.f, 1.f};

    for (int j0 = 0; j0 < M_; j0 += 8) {
        const int cb = j0 + cshift;
        v4f pv = *(const v4f*)(pred    + base  + cb);
        v4f gv = *(const v4f*)(gt      + base  + cb);
        v4f cg = *(const v4f*)(col_gt  + cstat + cb);
        v4f cc = *(const v4f*)(col_cnt + cstat + cb);

        float sn[4], sp[4], tn[4];
        #pragma unroll
        for (int k = 0; k < 4; ++k) {
            const float m  = (cb + k < tgt) ? rowv : 0.f;
            const float p  = fminf(fmaxf(pv[k], 0.f), 1.f) * m;
            const float g  = gv[k] * m;
            const float gp = p * g;
            // att_src = ((pred >= row_gt - beta) & mask) * row_cnt
            const float as = (m != 0.f && p >= rg - beta) ? rc : 0.f;
            float t = (as - g) * p;
            sn[k] = t * t;
            sp[k] = gp * gp;
            // att_tgt = ((pred >= col_gt - beta) & mask) * col_cnt
            const float at = (m != 0.f && p >= cg[k] - beta) ? cc[k] : 0.f;
            t = (at - g) * p;
            tn[k] = t * t;
        }

        c_sn = wmma_rowacc((v2f){sn[0], sn[1]}, ones, c_sn);
        c_sn = wmma_rowacc((v2f){sn[2], sn[3]}, ones, c_sn);
        c_sp = wmma_rowacc((v2f){sp[0], sp[1]}, ones, c_sp);
        c_sp = wmma_rowacc((v2f){sp[2], sp[3]}, ones, c_sp);

        const float stn = quad_group_reduce(tn[0], tn[1], tn[2], tn[3], lane);
        if ((lane & 3) == 0) {
            const int vk  = ((lane >> 3) & 1) | (((lane >> 2) & 1) << 1);
            atomicAdd(&cacc_tn[cb + vk], stn);
        }
    }

    if (lane == 0 || lane == 16) {
        const size_t ro = (size_t)b * N_ + rowbase + ((lane >> 4) << 3);
        #pragma unroll
        for (int k = 0; k < 8; ++k) {
            src_neg[ro + k] = c_sn[k];
            src_pos[ro + k] = c_sp[k];
        }
    }

    __syncthreads();
    for (int i = tid; i < M_; i += blockDim.x)
        atomicAdd(&tgt_neg[(size_t)b * M_ + i], cacc_tn[i]);
}

// corr[b] = dot(tgt_neg[b,:], col_cnt[b,:])
__global__ __launch_bounds__(256)
void k_corr(const float* __restrict__ tgt_neg, const float* __restrict__ col_cnt,
            float* __restrict__ corr) {
    __shared__ float red[256];
    const int b = blockIdx.x;
    float acc = 0.f;
    for (int m = threadIdx.x; m < M_; m += 256)
        acc += tgt_neg[(size_t)b * M_ + m] * col_cnt[(size_t)b * M_ + m];
    const float t = block_reduce_sum(acc, red);
    if (threadIdx.x == 0) corr[b] = t;
}

// loss = -0.5 * sum_{b, i < src_ns[b]} log(src_pos / (1 + src_neg + corr[b])) / sum(src_ns)
__global__ __launch_bounds__(256)
void k_finish(const float* __restrict__ src_neg, const float* __restrict__ src_pos,
              const float* __restrict__ corr, const int* __restrict__ src_ns,
              float* __restrict__ out) {
    __shared__ float red[256];
    float acc = 0.f;
    for (int idx = threadIdx.x; idx < B_ * N_; idx += 256) {
        const int b = idx >> 11;          // / N_
        const int i = idx & (N_ - 1);     // % N_
        if (i < src_ns[b]) {
            const float den = 1.f + src_neg[idx] + corr[b];
            acc += logf(src_pos[idx] / den);
        }
    }
    const float total = block_reduce_sum(acc, red);
    if (threadIdx.x == 0) {
        float nsum = 0.f;
        for (int b = 0; b < B_; ++b) nsum += (float)src_ns[b];
        out[0] = -0.5f * total / nsum;
    }
}

extern "C" void kernel_launch(void* const* d_in, const int* in_sizes, int n_in,
                              void* d_out, int out_size, void* d_ws, size_t ws_size,
                              hipStream_t stream) {
    const float* pred   = (const float*)d_in[0];
    const float* gtp    = (const float*)d_in[1];
    const int*   src_ns = (const int*)d_in[2];
    const int*   tgt_ns = (const int*)d_in[3];
    const float* beta   = (const float*)d_in[4];
    float* out = (float*)d_out;

    float* ws = (float*)d_ws;
    const size_t BN = (size_t)B_ * N_;   // == B_*M_
    float* row_gt  = ws + 0 * BN;
    float* row_cnt = ws + 1 * BN;
    float* src_neg = ws + 2 * BN;
    float* src_pos = ws + 3 * BN;
    float* col_gt  = ws + 4 * BN;
    float* col_cnt = ws + 5 * BN;
    float* tgt_neg = ws + 6 * BN;
    float* corr    = ws + 7 * BN;

    // Zero only the atomically-accumulated regions (col_gt, col_cnt, tgt_neg).
    hipMemsetAsync(col_gt, 0, 3 * BN * sizeof(float), stream);

    dim3 grid(N_ / 128, B_);
    dim3 block(256);
    k_stats<<<grid, block, 0, stream>>>(pred, gtp, src_ns, tgt_ns,
                                        row_gt, row_cnt, col_gt, col_cnt);
    k_terms<<<grid, block, 0, stream>>>(pred, gtp, src_ns, tgt_ns, beta,
                                        row_gt, row_cnt, col_gt, col_cnt,
                                        src_neg, src_pos, tgt_neg);
    k_corr<<<dim3(B_), dim3(256), 0, stream>>>(tgt_neg, col_cnt, corr);
    k_finish<<<dim3(1), dim3(256), 0, stream>>>(src_neg, src_pos, corr, src_ns, out);
}